// NEG_loss_8718783611216
// MI455X (gfx1250) — compile-verified
//
#include <hip/hip_runtime.h>
#include <hip/hip_bf16.h>

// ---------------------------------------------------------------------------
// NEG (negative-sampling) loss for MI455X / gfx1250, wave32.
//
// Shapes (fixed by the reference):
//   W:            [1,000,000, 64] f32   (256 MB table)
//   input_label:  [4096] i32            (scatter targets)
//   pos_sample:   [4096, 20] i32
//   update_batch: [4096, 64] f32
//   noise_idx:    [81920, 8] i32
//
// Memory-bound: ~210 MB of random 256B-row gathers -> ~9us at 23.3 TB/s.
// Strategy: 16 pairs per wave32; lane i / lane i+16 each own half of pair
// i's rows via float4 loads; per-lane FMA partials are reduced across K by
// V_WMMA_F32_16X16X4_F32 against an all-ones B matrix (matrix pipe used as
// a 16-row-at-once cross-lane reducer in f32, matching reference precision).
// ---------------------------------------------------------------------------

constexpr int NUM_CLASSES = 1000000;
constexpr int EMBED       = 64;
constexpr int BATCH       = 4096;
constexpr int WINDOW      = 20;
constexpr int NSAMP       = 8;
constexpr int NPAIRS      = BATCH * WINDOW;   // 81920
constexpr int NGROUPS     = NPAIRS / 16;      // 5120 waves

typedef float v2f __attribute__((ext_vector_type(2)));
typedef float v8f __attribute__((ext_vector_type(8)));

// Stable log(sigmoid(x)) = min(x,0) - log1p(exp(-|x|))
__device__ __forceinline__ float log_sigmoid(float x) {
    return fminf(x, 0.0f) - log1pf(__expf(-fabsf(x)));
}

// Resolve scatter: row p of update_batch if class g was scattered, else W[g].
__device__ __forceinline__ const float* row_ptr(int g, const int* __restrict__ map,
                                                const float* __restrict__ upd,
                                                const float* __restrict__ W) {
    int p = map[g];
    return (p >= 0) ? (upd + (size_t)p * EMBED) : (W + (size_t)g * EMBED);
}

__global__ void __launch_bounds__(256) init_map_kernel(int* __restrict__ map,
                                                       float* __restrict__ out) {
    int i = blockIdx.x * blockDim.x + threadIdx.x;
    if (i < NUM_CLASSES) map[i] = -1;
    if (i == 0) *out = 0.0f;   // d_out is poisoned; atomics accumulate into it
}

__global__ void __launch_bounds__(256) scatter_map_kernel(const int* __restrict__ label,
                                                          int* __restrict__ map) {
    int i = blockIdx.x * blockDim.x + threadIdx.x;
    if (i < BATCH) map[label[i]] = i;
}

__global__ void __launch_bounds__(256) neg_loss_kernel(
    const int*   __restrict__ label,
    const int*   __restrict__ pos,      // [B*W] flat, row-major
    const float* __restrict__ upd,      // [B, 64]
    const int*   __restrict__ noise,    // [B*W, 8]
    const float* __restrict__ W,        // [1e6, 64]
    const int*   __restrict__ map,      // [1e6] inverse scatter map (d_ws)
    float*       __restrict__ out)
{
    const int lane  = threadIdx.x & 31;
    const int wave  = blockIdx.x * (blockDim.x >> 5) + (threadIdx.x >> 5);
    if (wave >= NGROUPS) return;        // uniform per wave; WMMA EXEC stays all-1s

    const int m    = lane & 15;         // pair-in-group (matrix row M)
    const int half = lane >> 4;         // 0: elems 0..31, 1: elems 32..63
    const int n    = wave * 16 + m;     // global pair id
    const int b    = n % BATCH;         // jnp.tile pairing: inp row = n % B

    // ---- gather inp half-row (32 f32 = 8 x b128 per lane) ----------------
    const float*  ibase = row_ptr(label[b], map, upd, W);
    const float4* i4    = (const float4*)ibase + half * 8;
    float4 inp[8];
#pragma unroll
    for (int c = 0; c < 8; ++c) inp[c] = i4[c];

    const v2f ones = {1.0f, 1.0f};      // B matrix = all ones -> row-sum reduce
    float acc = 0.0f;

    // ---- positive target + 8 noise targets, one WMMA reduction each ------
#pragma unroll
    for (int t = 0; t < 1 + NSAMP; ++t) {
        const int g = (t == 0) ? pos[n] : noise[(size_t)n * NSAMP + (t - 1)];
        const float4* t4 = (const float4*)row_ptr(g, map, upd, W) + half * 8;

        float pa = 0.0f, pb = 0.0f;     // 2 partials/lane -> A-matrix K slots
#pragma unroll
        for (int c = 0; c < 8; ++c) {
            float4 tv = t4[c];
            pa = fmaf(inp[c].x, tv.x, fmaf(inp[c].y, tv.y, pa));
            pb = fmaf(inp[c].z, tv.z, fmaf(inp[c].w, tv.w, pb));
        }

        // A(16x4): lane m holds K=0,1 ; lane m+16 holds K=2,3.
        // C[i][j] = sum_k A[i][k] -> dot(inp[i], target[i]) replicated over N.
        v2f a; a.x = pa; a.y = pb;
        v8f c = {};
        c = __builtin_amdgcn_wmma_f32_16x16x4_f32(
                /*neg_a=*/false, a, /*neg_b=*/false, ones,
                /*c_mod=*/(short)0, c, /*reuse_a=*/false, /*reuse_b=*/false);

        // C VGPR r: lanes 0-15 -> row r, lanes 16-31 -> row r+8.
        // Noise score = dot(-noise_row, inp) => negate before log-sigmoid.
        const float sgn = (t == 0) ? 1.0f : -1.0f;
        float s = 0.0f;
#pragma unroll
        for (int r = 0; r < 8; ++r) s += log_sigmoid(sgn * c[r]);
        acc += s;
    }

    // lanes 0-15 hold sum over rows 0-7, lanes 16-31 over rows 8-15 (each
    // replicated 16x within its half); one xor-16 merge gives the group total.
    acc += __shfl_xor(acc, 16, 32);
    if (lane == 0)
        atomicAdd(out, -acc * (1.0f / (float)BATCH));
}

extern "C" void kernel_launch(void* const* d_in, const int* in_sizes, int n_in,
                              void* d_out, int out_size, void* d_ws, size_t ws_size,
                              hipStream_t stream) {
    const int*   label = (const int*)  d_in[0];
    const int*   pos   = (const int*)  d_in[1];
    const float* upd   = (const float*)d_in[2];
    const int*   noise = (const int*)  d_in[3];
    const float* W     = (const float*)d_in[4];
    // d_in[5] = num_sampled (compile-time 8 here)

    int*   map = (int*)d_ws;        // 4 MB inverse scatter map
    float* out = (float*)d_out;

    init_map_kernel<<<(NUM_CLASSES + 255) / 256, 256, 0, stream>>>(map, out);
    scatter_map_kernel<<<(BATCH + 255) / 256, 256, 0, stream>>>(label, map);

    // 5120 waves, 8 waves (256 threads) per block -> 640 blocks, no tail.
    neg_loss_kernel<<<NGROUPS / 8, 256, 0, stream>>>(label, pos, upd, noise, W, map, out);
}